// SelfNystromAttention_40561671144062
// MI455X (gfx1250) — compile-verified
//
#include <hip/hip_runtime.h>

typedef __attribute__((ext_vector_type(2))) float v2f;
typedef __attribute__((ext_vector_type(8))) float v8f;

#define WDIM   512
#define NSEQ   4096
#define BSZ    4
#define HEADS  8
#define DHEAD  64
#define LM     256     // landmarks
#define BH     32      // BSZ*HEADS
#define KCONV  33

// ---------------------------------------------------------------------------
// WMMA helpers: V_WMMA_F32_16X16X4_F32  (full-precision fp32 matrix core op)
// A 16x4 : lanes 0-15 -> M=lane, {v.x=K0, v.y=K1}; lanes 16-31 -> {K2,K3}
// B 4x16 : lanes 0-15 -> N=lane, {v.x=K0, v.y=K1}; lanes 16-31 -> {K2,K3}
// C/D    : VGPR i -> (M=i, N=lane) for lanes 0-15; (M=i+8, N=lane-16) hi half
// ---------------------------------------------------------------------------
__device__ inline v8f wmma4(v2f a, v2f b, v8f c) {
  return __builtin_amdgcn_wmma_f32_16x16x4_f32(
      /*neg_a=*/false, a, /*neg_b=*/false, b,
      /*c_mod=*/(short)0, c, /*reuse_a=*/false, /*reuse_b=*/false);
}

__device__ inline v2f load_a_frag(const float* __restrict__ A, int lda,
                                  int r0, int k0, int lane) {
  int m  = lane & 15;
  int kk = (lane >> 4) * 2;
  const float* p = A + (size_t)(r0 + m) * lda + k0 + kk;
  return v2f{p[0], p[1]};
}

__device__ inline v2f load_b_frag(const float* __restrict__ B, int ldb,
                                  int k0, int c0, int lane) {
  int n  = lane & 15;
  int kk = (lane >> 4) * 2;
  const float* p = B + (size_t)(k0 + kk) * ldb + c0 + n;
  return v2f{p[0], p[ldb]};
}

// B = Kmat^T where Kmat is (N x K) row-major  ->  B[k][n] = Kmat[n][k]
__device__ inline v2f load_bt_frag(const float* __restrict__ Km, int ldk,
                                   int k0, int c0, int lane) {
  int n  = lane & 15;
  int kk = (lane >> 4) * 2;
  const float* p = Km + (size_t)(c0 + n) * ldk + k0 + kk;
  return v2f{p[0], p[1]};
}

__device__ inline void store_tile(float* __restrict__ D, int ldd, int r, int c,
                                  int lane, v8f acc, float alpha, float diag) {
  int n = lane & 15, mhi = (lane >> 4) * 8;
#pragma unroll
  for (int i = 0; i < 8; ++i) {
    int rr = r + mhi + i, cc = c + n;
    float v = alpha * acc[i];
    if (rr == cc) v += diag;
    D[(size_t)rr * ldd + cc] = v;
  }
}

// ---------------------------------------------------------------------------
// reductions
// ---------------------------------------------------------------------------
__device__ inline float wave_sum(float v) {
#pragma unroll
  for (int o = 16; o > 0; o >>= 1) v += __shfl_xor(v, o, 32);
  return v;
}
__device__ inline float wave_max(float v) {
#pragma unroll
  for (int o = 16; o > 0; o >>= 1) v = fmaxf(v, __shfl_xor(v, o, 32));
  return v;
}
__device__ inline float block_sum(float v) {
  __shared__ float red[8];
  int wid = threadIdx.x >> 5, lane = threadIdx.x & 31;
  v = wave_sum(v);
  __syncthreads();
  if (lane == 0) red[wid] = v;
  __syncthreads();
  float r = 0.f;
  int nw = blockDim.x >> 5;
  for (int i = 0; i < nw; ++i) r += red[i];
  return r;
}
__device__ inline float block_max(float v) {
  __shared__ float red[8];
  int wid = threadIdx.x >> 5, lane = threadIdx.x & 31;
  v = wave_max(v);
  __syncthreads();
  if (lane == 0) red[wid] = v;
  __syncthreads();
  float r = -1e30f;
  int nw = blockDim.x >> 5;
  for (int i = 0; i < nw; ++i) r = fmaxf(r, red[i]);
  return r;
}

// ---------------------------------------------------------------------------
// LayerNorm: one block (256 thr) per row of 512
// ---------------------------------------------------------------------------
__global__ void ln_kernel(const float* __restrict__ x,
                          const float* __restrict__ g,
                          const float* __restrict__ b,
                          float* __restrict__ xn) {
  size_t row = blockIdx.x;
  const float* xr = x + row * WDIM;
  float* outr = xn + row * WDIM;
  int t = threadIdx.x;
  float v0 = xr[t], v1 = xr[t + 256];
  float s  = block_sum(v0 + v1);
  float s2 = block_sum(v0 * v0 + v1 * v1);
  float mu = s * (1.0f / WDIM);
  float var = s2 * (1.0f / WDIM) - mu * mu;
  float rstd = rsqrtf(var + 1e-5f);
  outr[t]       = (v0 - mu) * rstd * g[t]       + b[t];
  outr[t + 256] = (v1 - mu) * rstd * g[t + 256] + b[t + 256];
}

// ---------------------------------------------------------------------------
// qkv GEMM: (16384 x 512) @ (512 x 1536) scattered into (bh,n,d) head layout
// q pre-scaled by DHEAD^-0.5 = 0.125
// ---------------------------------------------------------------------------
__global__ void qkv_gemm_kernel(const float* __restrict__ A,   // xn
                                const float* __restrict__ B,   // w_qkv
                                float* __restrict__ q, float* __restrict__ k,
                                float* __restrict__ v) {
  int wave = threadIdx.x >> 5, lane = threadIdx.x & 31;
  int r0 = blockIdx.x * 128 + (wave >> 1) * 32;
  int c0 = blockIdx.y * 64 + (wave & 1) * 32;
  v8f a00 = {}, a01 = {}, a10 = {}, a11 = {};
#pragma unroll 4
  for (int kk = 0; kk < WDIM; kk += 4) {
    v2f fa0 = load_a_frag(A, WDIM, r0, kk, lane);
    v2f fa1 = load_a_frag(A, WDIM, r0 + 16, kk, lane);
    v2f fb0 = load_b_frag(B, 1536, kk, c0, lane);
    v2f fb1 = load_b_frag(B, 1536, kk, c0 + 16, lane);
    a00 = wmma4(fa0, fb0, a00); a01 = wmma4(fa0, fb1, a01);
    a10 = wmma4(fa1, fb0, a10); a11 = wmma4(fa1, fb1, a11);
  }
  int n = lane & 15, mhi = (lane >> 4) * 8;
#pragma unroll
  for (int ti = 0; ti < 2; ++ti)
#pragma unroll
    for (int tj = 0; tj < 2; ++tj) {
      v8f acc = (ti == 0) ? (tj == 0 ? a00 : a01) : (tj == 0 ? a10 : a11);
#pragma unroll
      for (int i = 0; i < 8; ++i) {
        int rr = r0 + ti * 16 + mhi + i;
        int cc = c0 + tj * 16 + n;
        int sel = cc >> 9;            // 0=q 1=k 2=v
        int rem = cc & 511;
        int h = rem >> 6, d = rem & 63;
        int bb = rr >> 12, nn = rr & 4095;
        size_t off = ((((size_t)bb * HEADS + h) * NSEQ) + nn) * DHEAD + d;
        float val = acc[i];
        if (sel == 0)      q[off] = val * 0.125f;
        else if (sel == 1) k[off] = val;
        else               v[off] = val;
      }
    }
}

// ---------------------------------------------------------------------------
// landmark means: mean over l=16 tokens.  blockIdx.y: 0 -> q_l, 1 -> k_l
// ---------------------------------------------------------------------------
__global__ void landmark_kernel(const float* __restrict__ q,
                                const float* __restrict__ k,
                                float* __restrict__ ql, float* __restrict__ kl) {
  int bh = blockIdx.x >> 8, j = blockIdx.x & 255, d = threadIdx.x;
  const float* src = (blockIdx.y == 0 ? q : k) +
                     (((size_t)bh * NSEQ) + (size_t)j * 16) * DHEAD + d;
  float s = 0.f;
#pragma unroll
  for (int t = 0; t < 16; ++t) s += src[(size_t)t * DHEAD];
  float* dst = (blockIdx.y == 0 ? ql : kl);
  dst[((size_t)bh * LM + j) * DHEAD + d] = s * (1.0f / 16.0f);
}

// ---------------------------------------------------------------------------
// logits = A @ Kmat^T  (batched, K=64)
// ---------------------------------------------------------------------------
__global__ void gemm_nt_kernel(const float* __restrict__ A,
                               const float* __restrict__ Km,
                               float* __restrict__ D, int K, int lda, int ldk,
                               int ldd, long long sA, long long sK,
                               long long sD) {
  A += (size_t)blockIdx.z * sA;
  Km += (size_t)blockIdx.z * sK;
  D += (size_t)blockIdx.z * sD;
  int wave = threadIdx.x >> 5, lane = threadIdx.x & 31;
  int r0 = blockIdx.x * 128 + (wave >> 1) * 32;
  int c0 = blockIdx.y * 64 + (wave & 1) * 32;
  v8f a00 = {}, a01 = {}, a10 = {}, a11 = {};
#pragma unroll 4
  for (int kk = 0; kk < K; kk += 4) {
    v2f fa0 = load_a_frag(A, lda, r0, kk, lane);
    v2f fa1 = load_a_frag(A, lda, r0 + 16, kk, lane);
    v2f fb0 = load_bt_frag(Km, ldk, kk, c0, lane);
    v2f fb1 = load_bt_frag(Km, ldk, kk, c0 + 16, lane);
    a00 = wmma4(fa0, fb0, a00); a01 = wmma4(fa0, fb1, a01);
    a10 = wmma4(fa1, fb0, a10); a11 = wmma4(fa1, fb1, a11);
  }
  store_tile(D, ldd, r0, c0, lane, a00, 1.f, 0.f);
  store_tile(D, ldd, r0, c0 + 16, lane, a01, 1.f, 0.f);
  store_tile(D, ldd, r0 + 16, c0, lane, a10, 1.f, 0.f);
  store_tile(D, ldd, r0 + 16, c0 + 16, lane, a11, 1.f, 0.f);
}

// ---------------------------------------------------------------------------
// generic batched GEMM:  D = alpha*(A@B) + diag*I
// ---------------------------------------------------------------------------
__global__ void gemm_nn_kernel(const float* __restrict__ A,
                               const float* __restrict__ B,
                               float* __restrict__ D, int K, int lda, int ldb,
                               int ldd, long long sA, long long sB,
                               long long sD, float alpha, float diag) {
  A += (size_t)blockIdx.z * sA;
  B += (size_t)blockIdx.z * sB;
  D += (size_t)blockIdx.z * sD;
  int wave = threadIdx.x >> 5, lane = threadIdx.x & 31;
  int r0 = blockIdx.x * 128 + (wave >> 1) * 32;
  int c0 = blockIdx.y * 64 + (wave & 1) * 32;
  v8f a00 = {}, a01 = {}, a10 = {}, a11 = {};
#pragma unroll 4
  for (int kk = 0; kk < K; kk += 4) {
    v2f fa0 = load_a_frag(A, lda, r0, kk, lane);
    v2f fa1 = load_a_frag(A, lda, r0 + 16, kk, lane);
    v2f fb0 = load_b_frag(B, ldb, kk, c0, lane);
    v2f fb1 = load_b_frag(B, ldb, kk, c0 + 16, lane);
    a00 = wmma4(fa0, fb0, a00); a01 = wmma4(fa0, fb1, a01);
    a10 = wmma4(fa1, fb0, a10); a11 = wmma4(fa1, fb1, a11);
  }
  store_tile(D, ldd, r0, c0, lane, a00, alpha, diag);
  store_tile(D, ldd, r0, c0 + 16, lane, a01, alpha, diag);
  store_tile(D, ldd, r0 + 16, c0, lane, a10, alpha, diag);
  store_tile(D, ldd, r0 + 16, c0 + 16, lane, a11, alpha, diag);
}

// ---------------------------------------------------------------------------
// softmax in place, one block per row, len is a multiple of 256 (<= 4096)
// ---------------------------------------------------------------------------
__global__ void softmax_kernel(float* __restrict__ data, int len) {
  size_t row = blockIdx.x;
  float* p = data + row * (size_t)len;
  int t = threadIdx.x, chunks = len >> 8;
  float loc[16];
  float mx = -1e30f;
  for (int c = 0; c < chunks; ++c) {
    loc[c] = p[t + (c << 8)];
    mx = fmaxf(mx, loc[c]);
  }
  mx = block_max(mx);
  float s = 0.f;
  for (int c = 0; c < chunks; ++c) {
    loc[c] = __expf(loc[c] - mx);
    s += loc[c];
  }
  s = block_sum(s);
  float inv = 1.0f / s;
  for (int c = 0; c < chunks; ++c) p[t + (c << 8)] = loc[c] * inv;
}

// ---------------------------------------------------------------------------
// pinv init: per-bh max row-sum / col-sum, then global max -> 1/(mr*mc)
// (reference uses jnp.max over ALL batch entries: a single global scalar)
// ---------------------------------------------------------------------------
__global__ void pinv_stats_kernel(const float* __restrict__ att2,
                                  float* __restrict__ stats) {
  int bh = blockIdx.x, t = threadIdx.x;
  const float* X = att2 + (size_t)bh * LM * LM;
  float rs = 0.f, cs = 0.f;
  for (int j = 0; j < LM; ++j) {
    rs += fabsf(X[(size_t)t * LM + j]);
    cs += fabsf(X[(size_t)j * LM + t]);
  }
  float mr = block_max(rs);
  float mc = block_max(cs);
  if (t == 0) { stats[bh] = mr; stats[BH + bh] = mc; }
}

__global__ void pinv_scale_kernel(float* __restrict__ stats) {
  float mr = wave_max(stats[threadIdx.x]);
  float mc = wave_max(stats[BH + threadIdx.x]);
  if (threadIdx.x == 0) stats[2 * BH] = 1.0f / (mr * mc);
}

__global__ void transpose_scale_kernel(const float* __restrict__ att2,
                                       float* __restrict__ z,
                                       const float* __restrict__ stats) {
  int bh = blockIdx.y, i = blockIdx.x, j = threadIdx.x;
  float inv = stats[2 * BH];
  z[(size_t)bh * LM * LM + (size_t)i * LM + j] =
      att2[(size_t)bh * LM * LM + (size_t)j * LM + i] * inv;
}

// t1 = 7I - xz
__global__ void diag7_kernel(const float* __restrict__ xz,
                             float* __restrict__ t1) {
  size_t idx = (size_t)blockIdx.x * 256 + threadIdx.x;
  int ij = (int)(idx & (LM * LM - 1));
  int i = ij >> 8, j = ij & 255;
  t1[idx] = (i == j ? 7.0f : 0.0f) - xz[idx];
}

// ---------------------------------------------------------------------------
// depthwise residual conv along sequence, added into out_heads
// ---------------------------------------------------------------------------
__global__ void conv_add_kernel(float* __restrict__ outh,
                                const float* __restrict__ v,
                                const float* __restrict__ w) {
  size_t idx = (size_t)blockIdx.x * 256 + threadIdx.x;  // BH*NSEQ*DHEAD
  int d = (int)(idx & 63);
  size_t r = idx >> 6;
  int n = (int)(r & 4095);
  int bh = (int)(r >> 12);
  int h = bh & 7;
  const float* vb = v + ((size_t)bh * NSEQ) * DHEAD + d;
  float acc = 0.f;
#pragma unroll
  for (int t = 0; t < KCONV; ++t) {
    int nn = n + t - KCONV / 2;
    if (nn >= 0 && nn < NSEQ) acc += w[h * KCONV + t] * vb[(size_t)nn * DHEAD];
  }
  outh[idx] += acc;
}

// ---------------------------------------------------------------------------
// final: d_out = x + (unpermute(out_heads) @ w_out + b_out)
// A[r][c] with r=(b,n), c=h*64+d pulled from (bh,n,d) layout on the fly
// ---------------------------------------------------------------------------
__device__ inline v2f load_a_headmap(const float* __restrict__ outh, int r0,
                                     int k0, int lane) {
  int m = lane & 15, kk = (lane >> 4) * 2;
  int rr = r0 + m;
  int bb = rr >> 12, nn = rr & 4095;
  int kx = k0 + kk;
  int h = kx >> 6, d = kx & 63;
  const float* p =
      outh + ((((size_t)bb * HEADS + h) * NSEQ) + nn) * DHEAD + d;
  return v2f{p[0], p[1]};
}

__global__ void final_gemm_kernel(const float* __restrict__ outh,
                                  const float* __restrict__ Wo,
                                  const float* __restrict__ bo,
                                  const float* __restrict__ xin,
                                  float* __restrict__ out) {
  int wave = threadIdx.x >> 5, lane = threadIdx.x & 31;
  int r0 = blockIdx.x * 128 + (wave >> 1) * 32;
  int c0 = blockIdx.y * 64 + (wave & 1) * 32;
  v8f a00 = {}, a01 = {}, a10 = {}, a11 = {};
#pragma unroll 4
  for (int kk = 0; kk < WDIM; kk += 4) {
    v2f fa0 = load_a_headmap(outh, r0, kk, lane);
    v2f fa1 = load_a_headmap(outh, r0 + 16, kk, lane);
    v2f fb0 = load_b_frag(Wo, WDIM, kk, c0, lane);
    v2f fb1 = load_b_frag(Wo, WDIM, kk, c0 + 16, lane);
    a00 = wmma4(fa0, fb0, a00); a01 = wmma4(fa0, fb1, a01);
    a10 = wmma4(fa1, fb0, a10); a11 = wmma4(fa1, fb1, a11);
  }
  int n = lane & 15, mhi = (lane >> 4) * 8;
#pragma unroll
  for (int ti = 0; ti < 2; ++ti)
#pragma unroll
    for (int tj = 0; tj < 2; ++tj) {
      v8f acc = (ti == 0) ? (tj == 0 ? a00 : a01) : (tj == 0 ? a10 : a11);
#pragma unroll
      for (int i = 0; i < 8; ++i) {
        int rr = r0 + ti * 16 + mhi + i;
        int cc = c0 + tj * 16 + n;
        out[(size_t)rr * WDIM + cc] =
            xin[(size_t)rr * WDIM + cc] + acc[i] + bo[cc];
      }
    }
}

// ---------------------------------------------------------------------------
// host driver
// ---------------------------------------------------------------------------
extern "C" void kernel_launch(void* const* d_in, const int* in_sizes, int n_in,
                              void* d_out, int out_size, void* d_ws,
                              size_t ws_size, hipStream_t stream) {
  const float* x     = (const float*)d_in[0];
  const float* gamma = (const float*)d_in[1];
  const float* beta  = (const float*)d_in[2];
  const float* w_qkv = (const float*)d_in[3];
  const float* w_out = (const float*)d_in[4];
  const float* b_out = (const float*)d_in[5];
  const float* res_w = (const float*)d_in[6];
  float* out = (float*)d_out;

  float* f = (float*)d_ws;
  size_t o = 0;
  float* xn   = f + o; o += (size_t)BSZ * NSEQ * WDIM;   // 8.39M
  float* q    = f + o; o += (size_t)BH * NSEQ * DHEAD;   // 8.39M
  float* kbuf = f + o; o += (size_t)BH * NSEQ * DHEAD;
  float* vbuf = f + o; o += (size_t)BH * NSEQ * DHEAD;
  float* ql   = f + o; o += (size_t)BH * LM * DHEAD;
  float* kl   = f + o; o += (size_t)BH * LM * DHEAD;
  float* att1 = f + o; o += (size_t)BH * NSEQ * LM;      // 33.5M
  float* att3 = f + o; o += (size_t)BH * LM * NSEQ;      // 33.5M
  float* att2 = f + o; o += (size_t)BH * LM * LM;
  float* za   = f + o; o += (size_t)BH * LM * LM;
  float* zb   = f + o; o += (size_t)BH * LM * LM;
  float* xz   = f + o; o += (size_t)BH * LM * LM;
  float* t1   = f + o; o += (size_t)BH * LM * LM;
  float* t2   = f + o; o += (size_t)BH * LM * LM;
  float* Sb   = f + o; o += (size_t)BH * LM * DHEAD;
  float* Pb   = f + o; o += (size_t)BH * LM * DHEAD;
  float* outh = f + o; o += (size_t)BH * NSEQ * DHEAD;
  float* stats = f + o; o += 128;

  const long long sQ  = (long long)NSEQ * DHEAD;  // per-bh stride of q/k/v
  const long long sL  = (long long)LM * DHEAD;    // per-bh stride of q_l/k_l/S/P
  const long long sA1 = (long long)NSEQ * LM;     // attn1/attn3 stride
  const long long sZ  = (long long)LM * LM;       // 256x256 stride

  // 1. layernorm
  ln_kernel<<<BSZ * NSEQ, 256, 0, stream>>>(x, gamma, beta, xn);
  // 2. qkv projection -> head layout, q scaled
  qkv_gemm_kernel<<<dim3(128, 24), 256, 0, stream>>>(xn, w_qkv, q, kbuf, vbuf);
  // 3. landmark means
  landmark_kernel<<<dim3(BH * LM, 2), DHEAD, 0, stream>>>(q, kbuf, ql, kl);
  // 4. logits
  gemm_nt_kernel<<<dim3(2, 4, BH), 256, 0, stream>>>(ql, kl, att2, DHEAD,
      DHEAD, DHEAD, LM, sL, sL, sZ);
  gemm_nt_kernel<<<dim3(32, 4, BH), 256, 0, stream>>>(q, kl, att1, DHEAD,
      DHEAD, DHEAD, LM, sQ, sL, sA1);
  gemm_nt_kernel<<<dim3(2, 64, BH), 256, 0, stream>>>(ql, kbuf, att3, DHEAD,
      DHEAD, DHEAD, NSEQ, sL, sQ, sA1);
  // 5. softmaxes
  softmax_kernel<<<BH * NSEQ, 256, 0, stream>>>(att1, LM);
  softmax_kernel<<<BH * LM, 256, 0, stream>>>(att2, LM);
  softmax_kernel<<<BH * LM, 256, 0, stream>>>(att3, NSEQ);
  // 6. pinv init: z = att2^T / (global max rowsum * global max colsum)
  pinv_stats_kernel<<<BH, 256, 0, stream>>>(att2, stats);
  pinv_scale_kernel<<<1, 32, 0, stream>>>(stats);
  transpose_scale_kernel<<<dim3(LM, BH), 256, 0, stream>>>(att2, za, stats);
  // 7. Newton-Schulz iterations (6x)
  float* zc = za; float* zn = zb;
  for (int it = 0; it < 6; ++it) {
    gemm_nn_kernel<<<dim3(2, 4, BH), 256, 0, stream>>>(att2, zc, xz, LM, LM,
        LM, LM, sZ, sZ, sZ, 1.0f, 0.0f);
    diag7_kernel<<<BH * LM * LM / 256, 256, 0, stream>>>(xz, t1);
    gemm_nn_kernel<<<dim3(2, 4, BH), 256, 0, stream>>>(xz, t1, t2, LM, LM, LM,
        LM, sZ, sZ, sZ, -1.0f, 15.0f);
    gemm_nn_kernel<<<dim3(2, 4, BH), 256, 0, stream>>>(xz, t2, t1, LM, LM, LM,
        LM, sZ, sZ, sZ, -1.0f, 13.0f);
    gemm_nn_kernel<<<dim3(2, 4, BH), 256, 0, stream>>>(zc, t1, zn, LM, LM, LM,
        LM, sZ, sZ, sZ, 0.25f, 0.0f);
    float* tmp = zc; zc = zn; zn = tmp;
  }
  // 8. S = attn3 @ v   (256x4096 @ 4096x64)
  gemm_nn_kernel<<<dim3(2, 1, BH), 256, 0, stream>>>(att3, vbuf, Sb, NSEQ,
      NSEQ, DHEAD, DHEAD, sA1, sQ, sL, 1.0f, 0.0f);
  // 9. P = pinv @ S
  gemm_nn_kernel<<<dim3(2, 1, BH), 256, 0, stream>>>(zc, Sb, Pb, LM, LM,
      DHEAD, DHEAD, sZ, sL, sL, 1.0f, 0.0f);
  // 10. out_heads = attn1 @ P   (associativity: avoids attn1 @ pinv)
  gemm_nn_kernel<<<dim3(32, 1, BH), 256, 0, stream>>>(att1, Pb, outh, LM, LM,
      DHEAD, DHEAD, sA1, sL, sQ, 1.0f, 0.0f);
  // 11. depthwise residual conv added in place
  conv_add_kernel<<<BH * NSEQ * DHEAD / 256, 256, 0, stream>>>(outh, vbuf,
                                                               res_w);
  // 12. output projection + residual
  final_gemm_kernel<<<dim3(128, 8), 256, 0, stream>>>(outh, w_out, b_out, x,
                                                      out);
}